// RX2_2448131359098
// MI455X (gfx1250) — compile-verified
//
#include <hip/hip_runtime.h>

// RX rotations on a 20-qubit state, batch 16, via tensor-product group matmuls
// on CDNA5 matrix cores (V_WMMA_F32_16X16X4_F32).
//
// Math: the product of 20 commuting RX(theta_q) gates factors into 5 groups of
// 4 qubits. Each group is a 16x16 complex matrix U = R + i*I applied along one
// 4-bit slice of the state index:  element e = (hi << (P0+4)) | (j << P0) | lo,
// group g occupies bits [P0, P0+3] with P0 = 16 - 4g (qubit q has stride
// 2^(19-q), so group-bit beta corresponds to qubit 4g + (3-beta)).
//
// Each pass: Y(16x16) = U * X(16x16) per tile of 16 columns, done as chains of
// four V_WMMA_F32_16X16X4_F32 per real product. Complex mul = 4 real matmuls.
// Entries of U: prod of c_b=cos(th/2) (bit match) / s_b=sin(th/2) (mismatch),
// times (-i)^popcount(M^j):  k%4==0:+R  1:-I  2:-R  3:+I  (one of R/I is 0).
//
// Memory: state kept as interleaved float2 [B][S] in d_out; pass 0 reads the
// real phi and writes all of d_out, passes 1..4 update d_out in place (tiles
// are disjoint across lanes/waves/blocks, and loads are consumed by WMMA
// before stores issue). 128 MB state fits in the 192 MB L2, so passes 1..4 are
// mostly L2-resident; worst-case HBM traffic ~1.2 GB -> ~50 us at 23.3 TB/s.

typedef __attribute__((ext_vector_type(2))) float v2f;
typedef __attribute__((ext_vector_type(8))) float v8f;

#define NQ     20
#define NBATCH 16
#define SSIZE  (1u << NQ)          // 2^20 amplitudes per batch element

// Grid geometry: 65536 tiles total (16 batches x 4096 tiles). 1024 blocks x
// 8 waves = 8192 waves; each wave owns 8 consecutive tiles of one batch
// (512 waves per batch), so the per-batch U matrices are built once per wave.
#define WAVES_PER_BATCH 512
#define TILES_PER_WAVE  8

template <int P0, bool FIRST>
__global__ __launch_bounds__(256) void rx_group_pass(
    const float* __restrict__ phi,      // [B][S] real input (pass 0 only)
    const float* __restrict__ thetas,   // [B][20]
    float*       __restrict__ state)    // [B][S][2] interleaved re/im
{
    const int lane  = threadIdx.x & 31;
    const int wid   = (int)blockIdx.x * 8 + (threadIdx.x >> 5);
    const int b     = wid >> 9;               // batch (512 waves per batch)
    const int wtile = wid & (WAVES_PER_BATCH - 1);
    const int h     = lane >> 4;              // lane half (K/M offset select)
    const int n     = lane & 15;              // tile column / A-matrix row

    // ---- Build A operands (U = R + i*I) once per wave --------------------
    // Group index g from P0; angle for group-bit beta is qubit 4g + (3-beta).
    const int g = (16 - P0) >> 2;
    float cb[4], sb[4];
#pragma unroll
    for (int beta = 0; beta < 4; ++beta) {
        float th = 0.5f * thetas[b * NQ + 4 * g + (3 - beta)];
        cb[beta] = cosf(th);
        sb[beta] = sinf(th);
    }

    // A layout (16x4 f32): lane%16 = row M; VGPR r with lane-half h -> column
    // K = r + 2*h of the 4-wide slice; slice kc covers global cols 4*kc..4*kc+3.
    v2f aR[4], aI[4], amI[4];
    const int M = n;
#pragma unroll
    for (int kc = 0; kc < 4; ++kc) {
#pragma unroll
        for (int r = 0; r < 2; ++r) {
            const int j = 4 * kc + r + 2 * h;
            const int x = M ^ j;
            float prod = 1.0f;
#pragma unroll
            for (int beta = 0; beta < 4; ++beta)
                prod *= ((x >> beta) & 1) ? sb[beta] : cb[beta];
            const int k = __popc(x);
            const float Rv = (k & 1) ? 0.0f : ((k & 2) ? -prod : prod);
            const float Iv = (k & 1) ? ((k & 2) ? prod : -prod) : 0.0f;
            aR[kc][r]  = Rv;
            aI[kc][r]  = Iv;
            amI[kc][r] = -Iv;
        }
    }

    const unsigned LOMASK = (1u << P0) - 1u;  // P0==0 -> 0 (lo is empty)

#pragma unroll 1
    for (int t = 0; t < TILES_PER_WAVE; ++t) {
        const unsigned tile  = (unsigned)wtile * TILES_PER_WAVE + (unsigned)t;
        const unsigned nc    = tile * 16u + (unsigned)n;       // this lane's column
        const unsigned hi    = nc >> P0;
        const unsigned lo    = nc & LOMASK;
        const unsigned ebase = (hi << (P0 + 4)) | lo;          // element at j = 0
        const size_t   base  = (size_t)b * SSIZE + ebase;

        // ---- Load B operands (X rows): B layout mirrors A: VGPR r holds
        // row K = r + 2*h of the 4-wide slice, N = lane%16. For P0>=4 each
        // fixed-j row is 16 consecutive float2 (128B) across a lane half.
        v2f bRe[4], bIm[4];
        if (FIRST) {
            const float* src = phi + base;
#pragma unroll
            for (int kc = 0; kc < 4; ++kc)
#pragma unroll
                for (int r = 0; r < 2; ++r) {
                    const int j = 4 * kc + r + 2 * h;
                    bRe[kc][r] = src[(size_t)j << P0];
                }
        } else {
            const v2f* src = (const v2f*)state + base;
#pragma unroll
            for (int kc = 0; kc < 4; ++kc)
#pragma unroll
                for (int r = 0; r < 2; ++r) {
                    const int j = 4 * kc + r + 2 * h;
                    const v2f v = src[(size_t)j << P0];
                    bRe[kc][r] = v.x;
                    bIm[kc][r] = v.y;
                }
        }

        // ---- Matrix-core complex multiply -------------------------------
        // Yre = R*Xre - I*Xim ; Yim = I*Xre + R*Xim   (Xim = 0 on pass 0)
        v8f accRe = {};
        v8f accIm = {};
#pragma unroll
        for (int kc = 0; kc < 4; ++kc) {
            accRe = __builtin_amdgcn_wmma_f32_16x16x4_f32(
                false, aR[kc], false, bRe[kc], (short)0, accRe, false, false);
            accIm = __builtin_amdgcn_wmma_f32_16x16x4_f32(
                false, aI[kc], false, bRe[kc], (short)0, accIm, false, false);
        }
        if (!FIRST) {
#pragma unroll
            for (int kc = 0; kc < 4; ++kc) {
                accRe = __builtin_amdgcn_wmma_f32_16x16x4_f32(
                    false, amI[kc], false, bIm[kc], (short)0, accRe, false, false);
                accIm = __builtin_amdgcn_wmma_f32_16x16x4_f32(
                    false, aR[kc], false, bIm[kc], (short)0, accIm, false, false);
            }
        }

        // ---- Store D (C-layout: VGPR r -> row M = r + 8*h, N = lane%16) --
        v2f* dst = (v2f*)state + base;
#pragma unroll
        for (int r = 0; r < 8; ++r) {
            const int Mo = r + 8 * h;
            v2f o;
            o.x = accRe[r];
            o.y = accIm[r];
            dst[(size_t)Mo << P0] = o;
        }
    }
}

extern "C" void kernel_launch(void* const* d_in, const int* in_sizes, int n_in,
                              void* d_out, int out_size, void* d_ws, size_t ws_size,
                              hipStream_t stream) {
    (void)in_sizes; (void)n_in; (void)d_ws; (void)ws_size; (void)out_size;
    const float* phi    = (const float*)d_in[0];   // [16][2^20] f32
    const float* thetas = (const float*)d_in[1];   // [16][20]   f32
    float* out = (float*)d_out;                    // [16][2^20][2] f32

    const dim3 grid(1024), block(256);
    // Pass 0: qubits 0..3   (bits 19..16), real input -> complex state in d_out
    rx_group_pass<16, true ><<<grid, block, 0, stream>>>(phi, thetas, out);
    // Passes 1..4: in-place on d_out
    rx_group_pass<12, false><<<grid, block, 0, stream>>>(phi, thetas, out);
    rx_group_pass< 8, false><<<grid, block, 0, stream>>>(phi, thetas, out);
    rx_group_pass< 4, false><<<grid, block, 0, stream>>>(phi, thetas, out);
    rx_group_pass< 0, false><<<grid, block, 0, stream>>>(phi, thetas, out);
}